// GaussianEncoderMessagePassing_14319420964995
// MI455X (gfx1250) — compile-verified
//
#include <hip/hip_runtime.h>

// ---------------- problem constants (from reference) ----------------
#define NODES   50000
#define EDGES   800000
#define GRAPHS  256
#define FEAT    128
#define STATE   64
#define ROUNDS  4
#define MDIM    32

typedef float v2f __attribute__((ext_vector_type(2)));
typedef float v8f __attribute__((ext_vector_type(8)));

enum { EP_RELU = 0, EP_RES_RELU = 1, EP_LIN = 2, EP_STD = 3 };

// ---------------------------------------------------------------------------
// Wave-per-tile fp32 WMMA GEMM: out[M x NS] = epilogue(A[M x K] @ W[K x NS] + b)
//   EP_RELU     : relu(acc + b)
//   EP_RES_RELU : res + relu(acc + b)          (residual update, in-place ok)
//   EP_LIN      : acc + b                      (mean head)
//   EP_STD      : exp(0.5 * clip(acc + b))     (std head)
//
// A-frag (16x4 f32, 2 VGPRs): lanes 0-15 -> M=lane, K = k+{0,1};
//                             lanes16-31 -> M=lane-16, K = k+{2,3}
// B-frag (4x16 f32, 2 VGPRs): symmetric (lanes 0-15: K=k+{0,1}, N=lane)
// C/D   (16x16 f32, 8 VGPRs): VGPR i -> (M = i + 8*(lane>=16), N = lane&15)
// ---------------------------------------------------------------------------
template<int K, int NS, int EPI>
__global__ __launch_bounds__(256)
void wmma_gemm(const float* __restrict__ A, const float* __restrict__ W,
               const float* __restrict__ bias, const float* __restrict__ res,
               float* __restrict__ out, int Mrows)
{
    const int tilesN = NS / 16;
    const int tilesM = Mrows / 16;
    int wave = (blockIdx.x * blockDim.x + threadIdx.x) >> 5;
    if (wave >= tilesM * tilesN) return;            // wave-uniform: EXEC stays all-1 for WMMA
    int lane = threadIdx.x & 31;
    int tm = wave / tilesN;
    int tn = wave % tilesN;
    int hi  = lane >> 4;                            // 0: lanes 0-15, 1: lanes 16-31
    int l16 = lane & 15;

    const float* Arow = A + (size_t)(tm * 16 + l16) * K + hi * 2;  // this lane's A row, K-phase
    const float* Wcol = W + tn * 16 + l16;                         // this lane's B column

    v8f acc = {};
#pragma unroll 8
    for (int k = 0; k < K; k += 4) {
        v2f a, b;
        a.x = Arow[k];
        a.y = Arow[k + 1];
        b.x = Wcol[(size_t)(k + hi * 2)     * NS];
        b.y = Wcol[(size_t)(k + hi * 2 + 1) * NS];
        // 8 args: (neg_a, A, neg_b, B, c_mod, C, reuse_a, reuse_b)
        acc = __builtin_amdgcn_wmma_f32_16x16x4_f32(false, a, false, b,
                                                    (short)0, acc, false, false);
    }

    int col = tn * 16 + l16;
    float bv = bias[col];
#pragma unroll
    for (int i = 0; i < 8; ++i) {
        int row = tm * 16 + i + hi * 8;
        float v = acc[i] + bv;
        if (EPI == EP_RELU || EPI == EP_RES_RELU) v = fmaxf(v, 0.0f);
        if (EPI == EP_RES_RELU)                   v += res[(size_t)row * NS + col];
        if (EPI == EP_STD) v = __expf(0.5f * fminf(fmaxf(v, -20.0f), 2.0f));
        out[(size_t)row * NS + col] = v;
    }
}

// ---------------------------------------------------------------------------
__global__ __launch_bounds__(256)
void zero_kernel(float* __restrict__ p, int n)
{
    int i = blockIdx.x * 256 + threadIdx.x;
    if (i < n) p[i] = 0.0f;
}

// agg[dst[e]] += msg[src[e]]  — thread = (edge, 4-float chunk); b128 gather (L2-resident),
// fp32 global atomics resolve in L2.
__global__ __launch_bounds__(256)
void edge_scatter(const float* __restrict__ msg, const int* __restrict__ src,
                  const int* __restrict__ dst, float* __restrict__ agg)
{
    int tid = blockIdx.x * 256 + threadIdx.x;
    if (tid >= EDGES * 16) return;
    int e = tid >> 4;
    int c = (tid & 15) << 2;
    int s = src[e];
    int d = dst[e];
    const float4 v = *(const float4*)(msg + (size_t)s * STATE + c);
    float* p = agg + (size_t)d * STATE + c;
    atomicAdd(p + 0, v.x);
    atomicAdd(p + 1, v.y);
    atomicAdd(p + 2, v.z);
    atomicAdd(p + 3, v.w);
}

// graph_state[batch[n]] += state[n]
__global__ __launch_bounds__(256)
void pool_kernel(const float* __restrict__ state, const int* __restrict__ batch,
                 float* __restrict__ gs)
{
    int tid = blockIdx.x * 256 + threadIdx.x;
    if (tid >= NODES * STATE) return;
    int n = tid >> 6;
    int f = tid & (STATE - 1);
    atomicAdd(&gs[(size_t)batch[n] * STATE + f], state[tid]);
}

// ---------------------------------------------------------------------------
extern "C" void kernel_launch(void* const* d_in, const int* in_sizes, int n_in,
                              void* d_out, int out_size, void* d_ws, size_t ws_size,
                              hipStream_t stream)
{
    const float* x      = (const float*)d_in[0];
    const int*   eidx   = (const int*)  d_in[1];   // [2, E]: row0=src(gather), row1=dst(scatter)
    const int*   batch  = (const int*)  d_in[2];
    const float* in_W   = (const float*)d_in[3];
    const float* in_b   = (const float*)d_in[4];
    const float* msg_W  = (const float*)d_in[5];   // [R,64,64]
    const float* msg_b  = (const float*)d_in[6];   // [R,64]
    const float* upd_W  = (const float*)d_in[7];
    const float* upd_b  = (const float*)d_in[8];
    const float* mean_W = (const float*)d_in[9];
    const float* mean_b = (const float*)d_in[10];
    const float* lv_W   = (const float*)d_in[11];
    const float* lv_b   = (const float*)d_in[12];

    float* state = (float*)d_ws;                       // [N,64]  12.8 MB
    float* msg   = state + (size_t)NODES * STATE;      // [N,64]  12.8 MB
    float* agg   = msg   + (size_t)NODES * STATE;      // [N,64]  12.8 MB
    float* gs    = agg   + (size_t)NODES * STATE;      // [G,64]  64 KB

    const int* src = eidx;
    const int* dst = eidx + EDGES;

    const int nodeWaves  = (NODES / 16) * (STATE / 16);         // 12500 waves
    const int nodeBlocks = (nodeWaves * 32 + 255) / 256;        // 1563
    const int headWaves  = (GRAPHS / 16) * (MDIM / 16);         // 32 waves
    const int headBlocks = (headWaves * 32 + 255) / 256;        // 4
    const int nsBlocks   = (NODES * STATE + 255) / 256;         // 12500
    const int esBlocks   = (EDGES * 16 + 255) / 256;            // 50000

    // input net: state = relu(x @ in_W + in_b)
    wmma_gemm<FEAT, STATE, EP_RELU><<<nodeBlocks, 256, 0, stream>>>(
        x, in_W, in_b, nullptr, state, NODES);

    zero_kernel<<<(GRAPHS * STATE + 255) / 256, 256, 0, stream>>>(gs, GRAPHS * STATE);

    for (int r = 0; r < ROUNDS; ++r) {
        // message = relu(state @ msg_W[r] + msg_b[r])
        wmma_gemm<STATE, STATE, EP_RELU><<<nodeBlocks, 256, 0, stream>>>(
            state, msg_W + (size_t)r * STATE * STATE, msg_b + (size_t)r * STATE,
            nullptr, msg, NODES);
        // aggregated = segment_sum(message[src], dst)
        zero_kernel<<<nsBlocks, 256, 0, stream>>>(agg, NODES * STATE);
        edge_scatter<<<esBlocks, 256, 0, stream>>>(msg, src, dst, agg);
        // state = state + relu(aggregated @ upd_W[r] + upd_b[r])   (in-place)
        wmma_gemm<STATE, STATE, EP_RES_RELU><<<nodeBlocks, 256, 0, stream>>>(
            agg, upd_W + (size_t)r * STATE * STATE, upd_b + (size_t)r * STATE,
            state, state, NODES);
    }

    // graph pooling
    pool_kernel<<<nsBlocks, 256, 0, stream>>>(state, batch, gs);

    // heads: d_out = [mean (256x32) | std (256x32)]
    float* outF = (float*)d_out;
    wmma_gemm<STATE, MDIM, EP_LIN><<<headBlocks, 256, 0, stream>>>(
        gs, mean_W, mean_b, nullptr, outF, GRAPHS);
    wmma_gemm<STATE, MDIM, EP_STD><<<headBlocks, 256, 0, stream>>>(
        gs, lv_W, lv_b, nullptr, outF + GRAPHS * MDIM, GRAPHS);
}